// SDecoder_89309549953745
// MI455X (gfx1250) — compile-verified
//
#include <hip/hip_runtime.h>
#include <hip/hip_bf16.h>

namespace {

constexpr int kThreads      = 256;  // 8 wave32 waves per block
constexpr int kBlocksPerRow = 256;

// Monotonic float -> uint mapping (preserves total order for all finite floats),
// packed with ~index so that u64-max == (max value, then lowest index), matching
// jnp.argmax first-hit tie-breaking.
__device__ __forceinline__ unsigned long long pack_key(float f, unsigned int idx) {
  unsigned int u   = __float_as_uint(f);
  unsigned int ord = (u & 0x80000000u) ? ~u : (u | 0x80000000u);
  return ((unsigned long long)ord << 32) | (unsigned long long)(0xFFFFFFFFu - idx);
}

// wave32 shuffle reduce + LDS cross-wave reduce; result valid on lane0/wave0.
__device__ __forceinline__ unsigned long long block_reduce_max(unsigned long long best) {
#pragma unroll
  for (int off = 16; off > 0; off >>= 1) {
    unsigned long long other = __shfl_xor(best, off, 32);  // wave32!
    if (other > best) best = other;
  }
  __shared__ unsigned long long smem[kThreads / 32];
  const int lane = threadIdx.x & 31;
  const int wave = threadIdx.x >> 5;
  if (lane == 0) smem[wave] = best;
  __syncthreads();
  if (wave == 0) {
    best = (lane < (kThreads / 32)) ? smem[lane] : 0ull;
#pragma unroll
    for (int off = (kThreads / 32) / 2; off > 0; off >>= 1) {
      unsigned long long other = __shfl_xor(best, off, 32);
      if (other > best) best = other;
    }
  }
  return best;
}

// Stage 1: streaming b128 argmax partials. One row handled by kBlocksPerRow blocks.
// Hot loop runs on raw floats (cmp + cndmask, ~3 VALU/element) so we stay HBM-bound;
// the 64-bit ordered key is packed exactly once per thread after the loop.
__global__ void SDecoder_rowmax_partial(const float* __restrict__ cls,
                                        unsigned long long* __restrict__ partials,
                                        int n) {
  const int row = blockIdx.x / kBlocksPerRow;
  const int blk = blockIdx.x % kBlocksPerRow;
  const float4* __restrict__ src = (const float4*)(cls + (size_t)row * (size_t)n);
  const int n4     = n >> 2;
  const int stride = kBlocksPerRow * kThreads;

  float        best    = -__builtin_huge_valf();
  unsigned int bestI4  = 0u;   // float4 index of current best
  unsigned int bestOff = 0u;   // element offset within that float4

  for (int i = blk * kThreads + (int)threadIdx.x; i < n4; i += stride) {
    // keep the stream ahead in GL2 (emits global_prefetch_b8 on gfx1250)
    __builtin_prefetch((const void*)(src + i + 2 * stride), 0, 1);
    float4 v = src[i];  // GLOBAL_LOAD_B128

    // quad-local argmax, strict '>' keeps the earliest element on ties
    float        m01 = v.x;  unsigned int o01 = 0u;
    if (v.y > m01) { m01 = v.y; o01 = 1u; }
    float        m23 = v.z;  unsigned int o23 = 2u;
    if (v.w > m23) { m23 = v.w; o23 = 3u; }
    float        q   = m01;  unsigned int oq  = o01;
    if (m23 > q)  { q = m23; oq = o23; }

    if (q > best) { best = q; bestI4 = (unsigned int)i; bestOff = oq; }
  }

  unsigned long long key = pack_key(best, (bestI4 << 2) | bestOff);

  // scalar tail if n % 4 != 0 (n = 2,000,000 -> empty, but stay generic)
  if (blk == 0) {
    for (int i = (n4 << 2) + (int)threadIdx.x; i < n; i += kThreads) {
      unsigned long long k = pack_key(cls[(size_t)row * (size_t)n + i], (unsigned int)i);
      if (k > key) key = k;
    }
  }

  key = block_reduce_max(key);
  if (threadIdx.x == 0) partials[blockIdx.x] = key;
}

// Stage 2: reduce partials per row, gather winner's reg/anchor, decode box.
__global__ void SDecoder_finalize(const unsigned long long* __restrict__ partials,
                                  const float* __restrict__ reg,
                                  const float* __restrict__ anchors,
                                  float* __restrict__ out,
                                  int n, int batch) {
  const int row = blockIdx.x;
  unsigned long long best = 0ull;
  for (int i = (int)threadIdx.x; i < kBlocksPerRow; i += kThreads) {
    unsigned long long k = partials[row * kBlocksPerRow + i];
    if (k > best) best = k;
  }
  best = block_reduce_max(best);

  if (threadIdx.x == 0) {
    // unpack score (exact inverse of the ordered-uint mapping) and index
    unsigned int ord = (unsigned int)(best >> 32);
    unsigned int u   = (ord & 0x80000000u) ? (ord ^ 0x80000000u) : ~ord;
    float score      = __uint_as_float(u);
    unsigned int idx = 0xFFFFFFFFu - (unsigned int)(best & 0xFFFFFFFFu);

    const float* r = reg + ((size_t)row * (size_t)n + idx) * 6;
    const float* a = anchors + (size_t)idx * 6;

    float lo[3], hi[3];
#pragma unroll
    for (int k = 0; k < 3; ++k) {
      float asz = a[3 + k];
      float c   = fmaf(r[k], asz, a[k]);           // pred_ctr
      float h   = 0.5f * expf(r[3 + k]) * asz;     // 0.5 * pred_sz
      lo[k] = fmaxf(c - h, 0.0f);
      hi[k] = fminf(c + h, 255.0f);                // IMG_SIZE-1 = 255 per dim
    }

    out[row] = score;                // max_scores, flat first
    float* box = out + batch + row * 6;
    box[0] = lo[0]; box[1] = lo[1]; box[2] = lo[2];
    box[3] = hi[0]; box[4] = hi[1]; box[5] = hi[2];
  }
}

}  // namespace

extern "C" void kernel_launch(void* const* d_in, const int* in_sizes, int n_in,
                              void* d_out, int out_size, void* d_ws, size_t ws_size,
                              hipStream_t stream) {
  const float* cls     = (const float*)d_in[0];  // (B, N) f32
  const float* reg     = (const float*)d_in[1];  // (B, N, 6) f32
  const float* anchors = (const float*)d_in[2];  // (N, 6) f32
  float* out           = (float*)d_out;          // B scores + B*6 boxes

  const int n     = in_sizes[2] / 6;             // N
  const int batch = in_sizes[0] / n;             // B

  unsigned long long* partials = (unsigned long long*)d_ws;  // batch*kBlocksPerRow u64

  dim3 block(kThreads);
  SDecoder_rowmax_partial<<<dim3(batch * kBlocksPerRow), block, 0, stream>>>(cls, partials, n);
  SDecoder_finalize<<<dim3(batch), block, 0, stream>>>(partials, reg, anchors, out, n, batch);
}